// GATResClassifier_30124900614323
// MI455X (gfx1250) — compile-verified
//
#include <hip/hip_runtime.h>
#include <hip/hip_bf16.h>
#include <math.h>

// ---------------- problem constants (from reference) ----------------
#define NN   50000
#define EE   800000
#define FIN  64
#define NC   64
#define HH   2
#define HC   64
#define OUTD 6
#define GG   512
#define NEG_GAT 0.2f
#define NEG_ACT 0.01f
#define LN_EPS  1e-5f

typedef float v2f __attribute__((ext_vector_type(2)));
typedef float v8f __attribute__((ext_vector_type(8)));

// order-preserving float<->uint encoding for atomic max
__device__ __forceinline__ unsigned f2ord(float f) {
    unsigned u = __float_as_uint(f);
    return (u & 0x80000000u) ? ~u : (u | 0x80000000u);
}
__device__ __forceinline__ float ord2f(unsigned u) {
    return (u & 0x80000000u) ? __uint_as_float(u & 0x7FFFFFFFu) : __uint_as_float(~u);
}
#define ORD_NEG_INF 0x007FFFFFu   // f2ord(-inf)

// ---------------- fills ----------------
__global__ void fill_f32(float* p, float v, int n) {
    int t = blockIdx.x * blockDim.x + threadIdx.x;
    if (t < n) p[t] = v;
}
__global__ void fill_u32(unsigned* p, unsigned v, int n) {
    int t = blockIdx.x * blockDim.x + threadIdx.x;
    if (t < n) p[t] = v;
}

// ---------------- WMMA fp32 GEMM: C[rows,O] = A[rows,K] @ W[K,O] + bias ----------------
// one wave = one 16x16 output tile; block = one 16-row stripe, O/16 waves across columns.
// A tile staged in LDS with stride K+1 (== 1 mod 64 banks -> conflict free).
template<int K, int O, int ACT>
__global__ __launch_bounds__((O / 16) * 32)
void gemm_bias_wmma(const float* __restrict__ A, const float* __restrict__ W,
                    const float* __restrict__ bias, float* __restrict__ Cout) {
    __shared__ float As[16][K + 1];
    const int tid = threadIdx.x;
    const int i0  = blockIdx.x * 16;
    for (int idx = tid; idx < 16 * K; idx += (O / 16) * 32) {
        int r = idx / K, c = idx % K;
        As[r][c] = A[(size_t)(i0 + r) * K + c];
    }
    __syncthreads();

    const int lane = tid & 31;
    const int half = lane >> 4;       // ISA 16x4 fp32 A layout: lanes 16-31 hold K+2,K+3
    const int lr   = lane & 15;
    const int j0   = (tid >> 5) * 16;

    v8f acc = {};
#pragma unroll
    for (int kk = 0; kk < K; kk += 4) {
        v2f a, b;
        a.x = As[lr][kk + 2 * half];
        a.y = As[lr][kk + 2 * half + 1];
        b.x = W[(kk + 2 * half) * O + j0 + lr];
        b.y = W[(kk + 2 * half + 1) * O + j0 + lr];
        acc = __builtin_amdgcn_wmma_f32_16x16x4_f32(false, a, false, b,
                                                    (short)0, acc, false, false);
    }
    const float bcol = bias[j0 + lr];
#pragma unroll
    for (int r = 0; r < 8; ++r) {
        float v = acc[r] + bcol;
        if (ACT == 1) v = v > 0.f ? v : 0.f;
        Cout[(size_t)(i0 + half * 8 + r) * O + j0 + lr] = v;
    }
}

// ---------------- GATv2 edge kernels (wave per edge; C=64 per head) ----------------
// edges 0..EE-1 come from edge_index; edges EE..EE+NN-1 are the added self-loops.
template<int H>
__global__ void edge_score_kernel(const int* __restrict__ ei,
                                  const float* __restrict__ xl, const float* __restrict__ xr,
                                  const float* __restrict__ att,
                                  float* __restrict__ score, unsigned* __restrict__ mmax) {
    const int HPC = H * 64, VW = HPC / 32;
    int wave = (blockIdx.x * blockDim.x + threadIdx.x) >> 5;
    int lane = threadIdx.x & 31;
    if (wave >= EE + NN) return;
    int src, dst;
    if (wave < EE) { src = ei[wave]; dst = ei[EE + wave]; }
    else           { src = wave - EE; dst = src; }

    float partial = 0.f;
    int c0 = lane * VW;
#pragma unroll
    for (int i = 0; i < VW; ++i) {
        int ch = c0 + i;
        float v = xl[(size_t)src * HPC + ch] + xr[(size_t)dst * HPC + ch];
        v = v > 0.f ? v : NEG_GAT * v;           // leaky_relu(0.2)
        partial += att[ch] * v;
    }
    // reduce within the head's lane group (16 lanes if H==2, 32 if H==1)
#pragma unroll
    for (int m = (H == 2 ? 8 : 16); m >= 1; m >>= 1)
        partial += __shfl_xor(partial, m, 32);

    int head = (H == 2) ? (lane >> 4) : 0;
    if ((lane & 15) == 0 && (H == 2 || lane == 0)) {
        score[(size_t)wave * H + head] = partial;
        atomicMax(&mmax[dst * H + head], f2ord(partial));   // max includes masked edges (matches ref)
    }
}

template<int H>
__global__ void finalize_max_kernel(const unsigned* __restrict__ mmax, float* __restrict__ mfin) {
    int t = blockIdx.x * blockDim.x + threadIdx.x;
    if (t >= NN * H) return;
    unsigned u = mmax[t];
    mfin[t] = (u == ORD_NEG_INF) ? 0.f : ord2f(u);   // where(isfinite(m), m, 0)
}

template<int H>
__global__ void edge_exp_kernel(const int* __restrict__ ei, const float* __restrict__ mfin,
                                float* __restrict__ score, float* __restrict__ denom) {
    int t = blockIdx.x * blockDim.x + threadIdx.x;
    if (t >= (EE + NN) * H) return;
    int e = t / H, h = t - e * H;
    int dst; float valid = 1.f;
    if (e < EE) { int s = ei[e]; dst = ei[EE + e]; valid = (s != dst) ? 1.f : 0.f; }
    else        { dst = e - EE; }
    float a = __expf(score[t] - mfin[dst * H + h]) * valid;
    score[t] = a;
    atomicAdd(&denom[dst * H + h], a);
}

template<int H>
__global__ void edge_aggregate_kernel(const int* __restrict__ ei,
                                      const float* __restrict__ xl,
                                      const float* __restrict__ score,
                                      const float* __restrict__ denom,
                                      float* __restrict__ agg) {
    const int HPC = H * 64, VW = HPC / 32;
    int wave = (blockIdx.x * blockDim.x + threadIdx.x) >> 5;
    int lane = threadIdx.x & 31;
    if (wave >= EE + NN) return;
    int src, dst;
    if (wave < EE) { src = ei[wave]; dst = ei[EE + wave]; }
    else           { src = wave - EE; dst = src; }
    int head = (H == 2) ? (lane >> 4) : 0;
    float a = score[(size_t)wave * H + head];
    if (a == 0.f) return;                       // masked self-loop duplicates contribute nothing
    float alpha = a / denom[dst * H + head];
    int c0 = lane * VW;
#pragma unroll
    for (int i = 0; i < VW; ++i) {
        int ch = c0 + i;
        atomicAdd(&agg[(size_t)dst * HPC + ch], alpha * xl[(size_t)src * HPC + ch]);
    }
}

// ---------------- bias (+residual) add fused with graph-LN reduction ----------------
__global__ void finish_conv_kernel(const float* __restrict__ agg, const float* __restrict__ bias,
                                   const float* __restrict__ res, int hasRes,
                                   float* __restrict__ y, float* __restrict__ stats,
                                   int n, int cmask) {
    float s = 0.f, sq = 0.f;
    for (int t = blockIdx.x * blockDim.x + threadIdx.x; t < n; t += gridDim.x * blockDim.x) {
        float v = agg[t] + bias[t & cmask];
        if (hasRes) v += res[t];
        y[t] = v;
        s += v; sq += v * v;
    }
#pragma unroll
    for (int m = 16; m >= 1; m >>= 1) { s += __shfl_xor(s, m, 32); sq += __shfl_xor(sq, m, 32); }
    __shared__ float sh[2][8];
    int wid = threadIdx.x >> 5, lane = threadIdx.x & 31;
    if (lane == 0) { sh[0][wid] = s; sh[1][wid] = sq; }
    __syncthreads();
    if (threadIdx.x == 0) {
        float ts = 0.f, tq = 0.f;
        for (int i = 0; i < (int)(blockDim.x >> 5); ++i) { ts += sh[0][i]; tq += sh[1][i]; }
        atomicAdd(&stats[0], ts);
        atomicAdd(&stats[1], tq);
    }
}

// graph-LN apply + leaky_relu(0.01)
__global__ void apply_ln_kernel(const float* __restrict__ y, const float* __restrict__ stats,
                                const float* __restrict__ w, const float* __restrict__ b,
                                float* __restrict__ out, int n, int cmask) {
    int t = blockIdx.x * blockDim.x + threadIdx.x;
    if (t >= n) return;
    float invM = 1.f / (float)n;
    float mean = stats[0] * invM;
    float var  = stats[1] * invM - mean * mean;
    float sd   = sqrtf(var > 0.f ? var : 0.f);
    float inv  = 1.f / (sd + LN_EPS);
    int c = t & cmask;
    float v = (y[t] - mean) * inv * w[c] + b[c];
    out[t] = v > 0.f ? v : NEG_ACT * v;
}

// ---------------- readout ----------------
__global__ void pool_kernel(const float* __restrict__ h, const int* __restrict__ batch,
                            float* __restrict__ pool, float* __restrict__ cnt) {
    int t = blockIdx.x * blockDim.x + threadIdx.x;
    if (t >= NN * NC) return;
    int nidx = t >> 6, c = t & 63;
    int g = batch[nidx];
    atomicAdd(&pool[g * NC + c], h[t]);
    if (c == 0) atomicAdd(&cnt[g], 1.f);
}
__global__ void pool_div_kernel(float* __restrict__ pool, const float* __restrict__ cnt) {
    int t = blockIdx.x * blockDim.x + threadIdx.x;
    if (t >= GG * NC) return;
    float c = cnt[t >> 6];
    pool[t] /= (c > 1.f ? c : 1.f);
}
__global__ void lin2_kernel(const float* __restrict__ r1, const float* __restrict__ W2,
                            const float* __restrict__ b2, float* __restrict__ out) {
    int t = blockIdx.x * blockDim.x + threadIdx.x;
    if (t >= GG * OUTD) return;
    int g = t / OUTD, o = t - g * OUTD;
    float s = b2[o];
#pragma unroll
    for (int c = 0; c < NC; ++c) s += r1[g * NC + c] * W2[c * OUTD + o];
    out[t] = s;
}

// ---------------- host-side conv driver ----------------
template<int H_, int KIN_>
static void run_conv(const float* Xin,
                     const float* Wl, const float* bl,
                     const float* Wr, const float* br,
                     const float* att, const float* bias,
                     const float* lnw, const float* lnb,
                     const float* resid, int hasRes,
                     float* xl, float* xr, float* agg,
                     float* score, unsigned* mmax, float* mfin, float* denom,
                     float* stats, const int* ei, float* outH, hipStream_t stream) {
    constexpr int O = H_ * 64;
    const int ETOT = EE + NN;
    gemm_bias_wmma<KIN_, O, 0><<<NN / 16, (O / 16) * 32, 0, stream>>>(Xin, Wl, bl, xl);
    gemm_bias_wmma<KIN_, O, 0><<<NN / 16, (O / 16) * 32, 0, stream>>>(Xin, Wr, br, xr);

    fill_u32<<<(NN * H_ + 255) / 256, 256, 0, stream>>>(mmax, ORD_NEG_INF, NN * H_);
    fill_f32<<<(NN * H_ + 255) / 256, 256, 0, stream>>>(denom, 0.f, NN * H_);
    fill_f32<<<(NN * O + 255) / 256, 256, 0, stream>>>(agg, 0.f, NN * O);

    edge_score_kernel<H_><<<(ETOT + 7) / 8, 256, 0, stream>>>(ei, xl, xr, att, score, mmax);
    finalize_max_kernel<H_><<<(NN * H_ + 255) / 256, 256, 0, stream>>>(mmax, mfin);
    edge_exp_kernel<H_><<<(ETOT * H_ + 255) / 256, 256, 0, stream>>>(ei, mfin, score, denom);
    edge_aggregate_kernel<H_><<<(ETOT + 7) / 8, 256, 0, stream>>>(ei, xl, score, denom, agg);

    fill_f32<<<1, 64, 0, stream>>>(stats, 0.f, 2);
    // t-buffer aliases xr (xr is dead after edge_score)
    finish_conv_kernel<<<1024, 256, 0, stream>>>(agg, bias, resid, hasRes, xr, stats, NN * O, O - 1);
    apply_ln_kernel<<<(NN * O + 255) / 256, 256, 0, stream>>>(xr, stats, lnw, lnb, outH, NN * O, O - 1);
}

extern "C" void kernel_launch(void* const* d_in, const int* in_sizes, int n_in,
                              void* d_out, int out_size, void* d_ws, size_t ws_size,
                              hipStream_t stream) {
    const float* x     = (const float*)d_in[0];
    const int*   ei    = (const int*)d_in[1];
    const int*   batch = (const int*)d_in[3];
    const float* W0 = (const float*)d_in[4];
    const float* b0 = (const float*)d_in[5];
    const float* W1 = (const float*)d_in[54];
    const float* b1 = (const float*)d_in[55];
    const float* W2 = (const float*)d_in[56];
    const float* b2 = (const float*)d_in[57];

    float* ws = (float*)d_ws;
    size_t o = 0;
    float*    bufA  = ws + o; o += (size_t)NN * 128;
    float*    bufB  = ws + o; o += (size_t)NN * 128;
    float*    xl    = ws + o; o += (size_t)NN * 128;
    float*    xr    = ws + o; o += (size_t)NN * 128;
    float*    agg   = ws + o; o += (size_t)NN * 128;
    float*    score = ws + o; o += (size_t)(EE + NN) * 2;
    unsigned* mmax  = (unsigned*)(ws + o); o += (size_t)NN * 2;
    float*    mfin  = ws + o; o += (size_t)NN * 2;
    float*    denom = ws + o; o += (size_t)NN * 2;
    float*    stats = ws + o; o += 16;
    float*    pool  = ws + o; o += (size_t)GG * NC;
    float*    cnt   = ws + o; o += GG;
    float*    r1    = ws + o; o += (size_t)GG * NC;

    // lin0: h = x @ W0 + b0   [N,64]
    gemm_bias_wmma<FIN, NC, 0><<<NN / 16, (NC / 16) * 32, 0, stream>>>(x, W0, b0, bufA);

    for (int blk = 0; blk < 3; ++blk) {
        int base = 6 + blk * 16;
        const float* c1Wl = (const float*)d_in[base + 0];
        const float* c1bl = (const float*)d_in[base + 1];
        const float* c1Wr = (const float*)d_in[base + 2];
        const float* c1br = (const float*)d_in[base + 3];
        const float* c1at = (const float*)d_in[base + 4];
        const float* c1bs = (const float*)d_in[base + 5];
        const float* l1w  = (const float*)d_in[base + 6];
        const float* l1b  = (const float*)d_in[base + 7];
        const float* c2Wl = (const float*)d_in[base + 8];
        const float* c2bl = (const float*)d_in[base + 9];
        const float* c2Wr = (const float*)d_in[base + 10];
        const float* c2br = (const float*)d_in[base + 11];
        const float* c2at = (const float*)d_in[base + 12];
        const float* c2bs = (const float*)d_in[base + 13];
        const float* l2w  = (const float*)d_in[base + 14];
        const float* l2b  = (const float*)d_in[base + 15];

        // conv1: [N,64] -> [N,128], H=2, no residual; output -> bufB
        run_conv<2, 64>(bufA, c1Wl, c1bl, c1Wr, c1br, c1at, c1bs, l1w, l1b,
                        bufA, 0, xl, xr, agg, score, mmax, mfin, denom, stats,
                        ei, bufB, stream);
        // conv2: [N,128] -> [N,64], H=1, residual = bufA (h0); output -> bufA
        run_conv<1, 128>(bufB, c2Wl, c2bl, c2Wr, c2br, c2at, c2bs, l2w, l2b,
                         bufA, 1, xl, xr, agg, score, mmax, mfin, denom, stats,
                         ei, bufA, stream);
    }

    // global mean pool
    fill_f32<<<(GG * NC + GG + 255) / 256, 256, 0, stream>>>(pool, 0.f, GG * NC + GG); // pool + cnt contiguous
    pool_kernel<<<(NN * NC + 255) / 256, 256, 0, stream>>>(bufA, batch, pool, cnt);
    pool_div_kernel<<<(GG * NC + 255) / 256, 256, 0, stream>>>(pool, cnt);

    // lin1 (relu) via WMMA, then lin2 scalar (OUT=6)
    gemm_bias_wmma<NC, NC, 1><<<GG / 16, (NC / 16) * 32, 0, stream>>>(pool, W1, b1, r1);
    lin2_kernel<<<(GG * OUTD + 255) / 256, 256, 0, stream>>>(r1, W2, b2, (float*)d_out);
}